// PatchMatch_87342454931714
// MI455X (gfx1250) — compile-verified
//
#include <hip/hip_runtime.h>
#include <hip/hip_bf16.h>

typedef __attribute__((ext_vector_type(2))) float v2f;
typedef __attribute__((ext_vector_type(8))) float v8f;

#define H      64
#define W      64
#define HW     4096
#define CCH    16
#define NB     4
#define KDIM   144            // 16 ch * 3 * 3
#define KEXT   148            // +1 for pn-fold, padded to multiple of 4
#define KP     (KEXT / 2)     // 74 float2 k-pairs
#define NTILE  (HW / 16)      // 256 tiles of 16 pixels

// ---------------------------------------------------------------------------
// Kernel 1: build extended patch descriptors in WMMA-native layout + qn.
// Layout: desc[((n*NTILE + tile)*KP + kp)*16 + pixInTile] : v2f = (k=2kp, k=2kp+1)
// Extended dims: q gets [1,0,0,0], p gets [-0.5*||p||^2, 0,0,0]  (pn fold).
// ---------------------------------------------------------------------------
__global__ void __launch_bounds__(256)
build_desc_kernel(const float* __restrict__ s, const float* __restrict__ t,
                  v2f* __restrict__ qdesc, v2f* __restrict__ pdesc,
                  float* __restrict__ qn) {
  const int which = blockIdx.y;                       // 0: s->q, 1: t->p
  const int gid   = blockIdx.x * blockDim.x + threadIdx.x;  // 0..16383
  const int n     = gid >> 12;
  const int pix   = gid & (HW - 1);
  const int y     = pix >> 6;
  const int x     = pix & (W - 1);
  const float* __restrict__ src = which ? t : s;

  auto val = [&](int k) -> float {
    const int ch  = k / 9;
    const int off = k - ch * 9;
    int yy = y + off / 3 - 1;
    int xx = x + off % 3 - 1;
    yy = yy < 0 ? 0 : (yy > H - 1 ? H - 1 : yy);      // replicate pad
    xx = xx < 0 ? 0 : (xx > W - 1 ? W - 1 : xx);
    return src[((n * CCH + ch) * H + yy) * W + xx];
  };

  float norm = 0.0f;
  for (int k = 0; k < KDIM; ++k) {
    const float v = val(k);
    norm += v * v;
  }

  const int tile = pix >> 4;
  const int pl   = pix & 15;
  v2f* __restrict__ dst =
      (which ? pdesc : qdesc) + ((size_t)(n * NTILE + tile) * KP) * 16 + pl;

  for (int kp = 0; kp < KDIM / 2; ++kp) {
    v2f e;
    e.x = val(2 * kp);
    e.y = val(2 * kp + 1);
    dst[kp * 16] = e;
  }
  v2f ext;                                            // k = 144,145
  ext.x = which ? (-0.5f * norm) : 1.0f;
  ext.y = 0.0f;
  dst[(KDIM / 2) * 16] = ext;
  v2f zz; zz.x = 0.0f; zz.y = 0.0f;                   // k = 146,147
  dst[(KDIM / 2 + 1) * 16] = zz;

  if (which == 0) qn[n * HW + pix] = norm;
}

// ---------------------------------------------------------------------------
// Kernel 2: one wave32 per (batch, query-tile). A = target tile (16xK, M dim),
// B = query tile (Kx16, N dim). WMMA f32 16x16x4, 37 K-steps, 2 accumulators.
// Per lane: column N = query (lane&15), 8 accum rows = targets M..M+7 with
// M-offset 8*(lane>>4). Running argmax of cross_ext == argmin of d2.
// ---------------------------------------------------------------------------
__global__ void __launch_bounds__(32)
nnf_wmma_kernel(const v2f* __restrict__ qdesc, const v2f* __restrict__ pdesc,
                const float* __restrict__ qn,
                int* __restrict__ outi, float* __restrict__ outf) {
  const int qt   = blockIdx.x;      // query tile 0..255
  const int n    = blockIdx.y;      // batch 0..3
  const int lane = threadIdx.x;     // 0..31
  const int kofs = lane >> 4;       // 0/1: which K-pair this half-wave owns
  const int pl   = lane & 15;

  __shared__ v2f qtile[KP * 16];    // 9.25 KB query descriptors
  {
    const v2f* __restrict__ qsrc = qdesc + (size_t)(n * NTILE + qt) * KP * 16;
    for (int i = lane; i < KP * 16; i += 32) qtile[i] = qsrc[i];  // 37 full iters
  }
  __syncthreads();

  float mv = -3.402823466e+38f;
  int   mi = 0;

  for (int tt = 0; tt < NTILE; ++tt) {
    const v2f* __restrict__ psrc = pdesc + (size_t)(n * NTILE + tt) * KP * 16;
    v8f c0 = {};
    v8f c1 = {};
#pragma unroll
    for (int k4 = 0; k4 < 36; k4 += 2) {
      const int kp0 = 2 * k4 + kofs;
      v2f a0 = psrc[kp0 * 16 + pl];
      v2f b0 = qtile[kp0 * 16 + pl];
      c0 = __builtin_amdgcn_wmma_f32_16x16x4_f32(false, a0, false, b0,
                                                 (short)0, c0, false, false);
      const int kp1 = 2 * (k4 + 1) + kofs;
      v2f a1 = psrc[kp1 * 16 + pl];
      v2f b1 = qtile[kp1 * 16 + pl];
      c1 = __builtin_amdgcn_wmma_f32_16x16x4_f32(false, a1, false, b1,
                                                 (short)0, c1, false, false);
    }
    { // 37th (final) K-step
      const int kp = 2 * 36 + kofs;
      v2f a = psrc[kp * 16 + pl];
      v2f b = qtile[kp * 16 + pl];
      c0 = __builtin_amdgcn_wmma_f32_16x16x4_f32(false, a, false, b,
                                                 (short)0, c0, false, false);
    }
    const int jbase = tt * 16 + 8 * kofs;
#pragma unroll
    for (int r = 0; r < 8; ++r) {
      const float v = c0[r] + c1[r];
      const int   j = jbase + r;
      if (v > mv) { mv = v; mi = j; }   // strict '>' keeps earliest index
    }
  }

  // Merge the two half-wave partials covering the same query column.
  const float ov = __shfl_xor(mv, 16, 32);
  const int   oi = __shfl_xor(mi, 16, 32);
  if (ov > mv || (ov == mv && oi < mi)) { mv = ov; mi = oi; }

  if (lane < 16) {
    const int   i    = qt * 16 + lane;          // flat query pixel
    const float nndv = qn[n * HW + i] - 2.0f * mv;
    const int   idy  = mi >> 6;
    const int   idx  = mi & (W - 1);
    outi[(n * 2 + 0) * HW + i] = idy;           // nnf[:,0]
    outi[(n * 2 + 1) * HW + i] = idx;           // nnf[:,1]
    outf[NB * 2 * HW + n * HW + i] = nndv;      // nnd after nnf block
  }
}

// ---------------------------------------------------------------------------
extern "C" void kernel_launch(void* const* d_in, const int* in_sizes, int n_in,
                              void* d_out, int out_size, void* d_ws, size_t ws_size,
                              hipStream_t stream) {
  const float* s = (const float*)d_in[0];
  const float* t = (const float*)d_in[1];
  // d_in[2] = patch_size (always 3 for this problem; layout hardcoded for it)

  const size_t descElems = (size_t)NB * NTILE * KP * 16;   // v2f elements
  v2f*   qdesc = (v2f*)d_ws;
  v2f*   pdesc = qdesc + descElems;
  float* qn    = (float*)(pdesc + descElems);
  // total workspace use: 2 * 9.70 MB + 64 KB  (~19.5 MB)

  build_desc_kernel<<<dim3(64, 2), 256, 0, stream>>>(s, t, qdesc, pdesc, qn);
  nnf_wmma_kernel<<<dim3(NTILE, NB), 32, 0, stream>>>(qdesc, pdesc, qn,
                                                      (int*)d_out, (float*)d_out);
}